// MemorySubsets_36507222016792
// MI455X (gfx1250) — compile-verified
//
#include <hip/hip_runtime.h>
#include <stdint.h>

// Problem constants (from reference): B=8, M=128, H=8, D=64, K=12
#define B_ 8
#define M_ 128
#define H_ 8
#define D_ 64
#define K_ 12
#define SLAB (H_ * D_ * D_)            // 32768 floats per (b,m) slab = 128 KiB
#define MAT_ELEMS (B_ * M_ * SLAB)     // 33,554,432
#define NORM_ELEMS (B_ * M_ * H_ * D_) // 524,288

// Copy-path tiling: 256 threads x 64 B = 16 KiB per chunk, 8 chunks per slab,
// double-buffered in LDS (2 x 16 KiB). Per-wave in flight: 4 loads + 4 stores
// of b128 = 4 KB between asynccnt drains.
#define CHUNK_BYTES 16384
#define NCHUNK (SLAB * 4 / CHUNK_BYTES)   // 8

// ---------------------------------------------------------------------------
// CDNA5 async global<->LDS DMA helpers (ASYNCcnt path, CDNA5 ISA)
// GVS addressing: mem = SGPR_base + VGPR_i32_offset  (offsets here < 2^27)
// ---------------------------------------------------------------------------
__device__ __forceinline__ void async_load_b128(uint32_t lds_addr, uint32_t voff,
                                                const void* sbase) {
  asm volatile("global_load_async_to_lds_b128 %0, %1, %2"
               :: "v"(lds_addr), "v"(voff), "s"(sbase) : "memory");
}
__device__ __forceinline__ void async_store_b128(uint32_t voff, uint32_t lds_addr,
                                                 void* sbase) {
  asm volatile("global_store_async_from_lds_b128 %0, %1, %2"
               :: "v"(voff), "v"(lds_addr), "s"(sbase) : "memory");
}
__device__ __forceinline__ void wait_async0() {
  asm volatile("s_wait_asynccnt 0" ::: "memory");
}

__device__ __forceinline__ float sigmoidf(float x) {
  return 1.0f / (1.0f + __expf(-x));
}

// ---------------------------------------------------------------------------
// Matrix kernel: one block per (b,m) slab (H*D*D = 32768 floats = 128 KiB).
//  - no selected k hits this m  -> pure streaming copy via async LDS DMA
//  - matched                    -> fused decay blend (decay shared by all
//    matching k since they hit the same m):
//        out = x + sigmoid(main[m,h,d1]+aux[m,d2]) * (sum_k p_k*upd_k - x*sum_k p_k)
// ---------------------------------------------------------------------------
__global__ __launch_bounds__(256) void mat_kernel(
    const float* __restrict__ matrix,   // (B,M,H,D,D)
    const float* __restrict__ mupd,     // (B,K,H,D,D)
    const float* __restrict__ mainlog,  // (M,H,D)
    const float* __restrict__ auxlog,   // (M,D)
    const int*   __restrict__ selidx,   // (B,K)
    const float* __restrict__ selp,     // (B,K)
    float* __restrict__ outm)           // (B,M,H,D,D)
{
  const int bm  = blockIdx.x;
  const int b   = bm >> 7;        // / M_
  const int m   = bm & (M_ - 1);  // % M_
  const int tid = threadIdx.x;

  __shared__ int    s_k[K_];
  __shared__ float  s_p[K_];
  __shared__ int    s_n;
  __shared__ float  s_main[H_ * D_];   // 2 KiB
  __shared__ float  s_aux[D_];         // 256 B
  __shared__ float4 s_buf[2][CHUNK_BYTES / 16];  // 2 x 16 KiB staging

  if (tid == 0) {
    int n = 0;
    for (int k = 0; k < K_; ++k) {
      if (selidx[b * K_ + k] == m) { s_k[n] = k; s_p[n] = selp[b * K_ + k]; ++n; }
    }
    s_n = n;
  }
  __syncthreads();
  const int nmatch = s_n;

  const size_t slab_elem = (size_t)bm * SLAB;

  if (nmatch == 0) {
    // -------- streaming copy: 8 chunks of 16 KiB, double-buffered async DMA
    const uint32_t base_byte = (uint32_t)(slab_elem * 4u);
    const uint32_t toff      = (uint32_t)tid * 64u;  // 4 x b128 per thread/chunk
    const uint32_t lds0 = (uint32_t)(uintptr_t)(&s_buf[0][0]) + toff;
    const uint32_t lds1 = (uint32_t)(uintptr_t)(&s_buf[1][0]) + toff;

    // prologue: load chunk 0 into buffer 0
    {
      const uint32_t g0 = base_byte + toff;
      async_load_b128(lds0,      g0,      matrix);
      async_load_b128(lds0 + 16, g0 + 16, matrix);
      async_load_b128(lds0 + 32, g0 + 32, matrix);
      async_load_b128(lds0 + 48, g0 + 48, matrix);
    }

    for (int c = 0; c < NCHUNK; ++c) {
      wait_async0();  // chunk c loaded; chunk c-1 store drained -> buffers safe
      const uint32_t lc = (c & 1) ? lds1 : lds0;
      const uint32_t go = base_byte + (uint32_t)c * CHUNK_BYTES + toff;
      async_store_b128(go,      lc,      outm);
      async_store_b128(go + 16, lc + 16, outm);
      async_store_b128(go + 32, lc + 32, outm);
      async_store_b128(go + 48, lc + 48, outm);
      if (c + 1 < NCHUNK) {
        const uint32_t ln = (c & 1) ? lds0 : lds1;
        const uint32_t gn = base_byte + (uint32_t)(c + 1) * CHUNK_BYTES + toff;
        async_load_b128(ln,      gn,      matrix);
        async_load_b128(ln + 16, gn + 16, matrix);
        async_load_b128(ln + 32, gn + 32, matrix);
        async_load_b128(ln + 48, gn + 48, matrix);
      }
    }
    wait_async0();
    return;
  }

  // -------- matched path: fused decay blend
  for (int i = tid; i < H_ * D_; i += 256) s_main[i] = mainlog[m * H_ * D_ + i];
  if (tid < D_) s_aux[tid] = auxlog[m * D_ + tid];
  __syncthreads();

  float psum = 0.0f;
  for (int j = 0; j < nmatch; ++j) psum += s_p[j];

  const float* __restrict__ xin  = matrix + slab_elem;
  float* __restrict__       xout = outm   + slab_elem;

  for (int it = 0; it < SLAB / 1024; ++it) {   // 32 iters x 256 threads x float4
    const int fi = it * 1024 + tid * 4;
    const int d2 = fi & (D_ - 1);
    const int d1 = (fi >> 6) & (D_ - 1);
    const int h  = fi >> 12;

    const float4 x = *(const float4*)(xin + fi);

    float4 u = make_float4(0.f, 0.f, 0.f, 0.f);
    for (int j = 0; j < nmatch; ++j) {
      const int   k = s_k[j];
      const float p = s_p[j];
      const float4 ud = *(const float4*)(
          mupd + ((((size_t)b * K_ + k) * H_ + h) * D_ + d1) * D_ + d2);
      u.x += p * ud.x; u.y += p * ud.y; u.z += p * ud.z; u.w += p * ud.w;
    }

    const float  ms = s_main[h * D_ + d1];
    const float4 av = *(const float4*)(&s_aux[d2]);
    float4 dec;
    dec.x = sigmoidf(ms + av.x);
    dec.y = sigmoidf(ms + av.y);
    dec.z = sigmoidf(ms + av.z);
    dec.w = sigmoidf(ms + av.w);

    float4 o;
    o.x = x.x + dec.x * (u.x - psum * x.x);
    o.y = x.y + dec.y * (u.y - psum * x.y);
    o.z = x.z + dec.z * (u.z - psum * x.z);
    o.w = x.w + dec.w * (u.w - psum * x.w);

    *(float4*)(xout + fi) = o;
  }
}

// ---------------------------------------------------------------------------
// Normalizer kernel: elementwise over (B,M,H,D), float4 along D.
//   out = n + sum_{k: idx==m} p_k * sigmoid(main[m,h,d]) * (nupd_k - n)
// ---------------------------------------------------------------------------
__global__ __launch_bounds__(256) void norm_kernel(
    const float* __restrict__ normalizer,  // (B,M,H,D)
    const float* __restrict__ nupd,        // (B,K,H,D)
    const float* __restrict__ mainlog,     // (M,H,D)
    const int*   __restrict__ selidx,      // (B,K)
    const float* __restrict__ selp,        // (B,K)
    float* __restrict__ outn)              // (B,M,H,D)
{
  const int gid = blockIdx.x * 256 + threadIdx.x;
  const int e   = gid * 4;
  if (e >= NORM_ELEMS) return;

  const int d  = e & (D_ - 1);
  const int hm = e >> 6;            // (b*M + m)*H + h
  const int h  = hm & (H_ - 1);
  const int bm = hm >> 3;
  const int b  = bm >> 7;
  const int m  = bm & (M_ - 1);

  const float4 n = *(const float4*)(normalizer + e);
  float4 acc = n;

  for (int k = 0; k < K_; ++k) {
    if (selidx[b * K_ + k] == m) {
      const float p = selp[b * K_ + k];
      const float4 nu = *(const float4*)(
          nupd + (((size_t)b * K_ + k) * H_ + h) * D_ + d);
      const float4 ms = *(const float4*)(mainlog + ((size_t)m * H_ + h) * D_ + d);
      acc.x += p * sigmoidf(ms.x) * (nu.x - n.x);
      acc.y += p * sigmoidf(ms.y) * (nu.y - n.y);
      acc.z += p * sigmoidf(ms.z) * (nu.z - n.z);
      acc.w += p * sigmoidf(ms.w) * (nu.w - n.w);
    }
  }
  *(float4*)(outn + e) = acc;
}

// ---------------------------------------------------------------------------
extern "C" void kernel_launch(void* const* d_in, const int* in_sizes, int n_in,
                              void* d_out, int out_size, void* d_ws, size_t ws_size,
                              hipStream_t stream) {
  const float* matrix     = (const float*)d_in[0];
  const float* normalizer = (const float*)d_in[1];
  const float* mupd       = (const float*)d_in[2];
  const float* nupd       = (const float*)d_in[3];
  const float* mainlog    = (const float*)d_in[4];
  const float* auxlog     = (const float*)d_in[5];
  const int*   selidx     = (const int*)d_in[6];
  const float* selp       = (const float*)d_in[7];

  float* outm = (float*)d_out;
  float* outn = outm + (size_t)MAT_ELEMS;

  mat_kernel<<<B_ * M_, 256, 0, stream>>>(matrix, mupd, mainlog, auxlog,
                                          selidx, selp, outm);
  norm_kernel<<<NORM_ELEMS / 1024, 256, 0, stream>>>(normalizer, nupd, mainlog,
                                                     selidx, selp, outn);
}